// Aggregator_67336497267219
// MI455X (gfx1250) — compile-verified
//
#include <hip/hip_runtime.h>
#include <hip/hip_bf16.h>

#define N_ENTITIES  100000
#define N_USERS     50000
#define CHANNEL     64
#define N_RELATIONS 16
#define N_FACTORS   4
#define N_EDGES     1600000
#define N_INTER     1000000

typedef float v2f __attribute__((ext_vector_type(2)));
typedef float v8f __attribute__((ext_vector_type(8)));

// ---------------------------------------------------------------------------
// Zero out d_out (entity + user accumulators) and the count buffer in ws.
// ---------------------------------------------------------------------------
__global__ void k_zero(float* __restrict__ out, float* __restrict__ cnt) {
    const int n_out = (N_ENTITIES + N_USERS) * CHANNEL;
    int idx = blockIdx.x * blockDim.x + threadIdx.x;
    if (idx < n_out)                       out[idx] = 0.f;
    else if (idx < n_out + N_ENTITIES)     cnt[idx - n_out] = 0.f;
}

// ---------------------------------------------------------------------------
// disen = softmax(disen_weight_att, axis=-1) @ weight    -> [4, 64] in ws
// One block, 256 threads: thread t handles (factor f = t/64, channel c = t%64)
// ---------------------------------------------------------------------------
__global__ void k_disen(const float* __restrict__ att,
                        const float* __restrict__ weight,
                        float* __restrict__ disen) {
    int t = threadIdx.x;
    int f = t >> 6, c = t & 63;
    float mx = -3.4e38f;
#pragma unroll
    for (int r = 0; r < N_RELATIONS; ++r) mx = fmaxf(mx, att[f * N_RELATIONS + r]);
    float s = 0.f;
#pragma unroll
    for (int r = 0; r < N_RELATIONS; ++r) s += __expf(att[f * N_RELATIONS + r] - mx);
    float inv = 1.f / s;
    float acc = 0.f;
#pragma unroll
    for (int r = 0; r < N_RELATIONS; ++r)
        acc += __expf(att[f * N_RELATIONS + r] - mx) * inv * weight[r * CHANNEL + c];
    disen[f * CHANNEL + c] = acc;
}

// ---------------------------------------------------------------------------
// Edge aggregation: 16 lanes per edge, float4 (4 channels) per lane.
// att = sigmoid(dot(ent[head], rel)); scatter att*ent[tail]*rel into out_ent.
// entity_emb (25.6MB) and out_ent (25.6MB) are L2-resident on MI455X (192MB).
// ---------------------------------------------------------------------------
__global__ void __launch_bounds__(256) k_edges(
        const float4* __restrict__ ent, const float4* __restrict__ wgt,
        const int* __restrict__ head, const int* __restrict__ tail,
        const int* __restrict__ etype,
        float* __restrict__ out_ent, float* __restrict__ cnt) {
    int tid = blockIdx.x * blockDim.x + threadIdx.x;
    int e   = tid >> 4;          // one 16-lane group per edge (exact: 1.6M*16/256 blocks)
    int sub = tid & 15;

    int h = head[e];
    int t = tail[e];
    int r = etype[e] - 1;        // edge_type is 1-indexed

    float4 hv = ent[h * 16 + sub];
    float4 rv = wgt[r * 16 + sub];
    float p = hv.x * rv.x + hv.y * rv.y + hv.z * rv.z + hv.w * rv.w;
    // reduce within the 16-lane group (xor masks < 16 stay inside the group)
    p += __shfl_xor(p, 1);
    p += __shfl_xor(p, 2);
    p += __shfl_xor(p, 4);
    p += __shfl_xor(p, 8);
    float attv = 1.f / (1.f + __expf(-p));

    float4 tv = ent[t * 16 + sub];
    int base = h * CHANNEL + sub * 4;
    atomicAdd(&out_ent[base + 0], attv * tv.x * rv.x);
    atomicAdd(&out_ent[base + 1], attv * tv.y * rv.y);
    atomicAdd(&out_ent[base + 2], attv * tv.z * rv.z);
    atomicAdd(&out_ent[base + 3], attv * tv.w * rv.w);
    if (sub == 0) atomicAdd(&cnt[h], 1.f);
}

// ---------------------------------------------------------------------------
// Interaction aggregation: user_agg += val * entity_emb[col] scattered by row.
// ---------------------------------------------------------------------------
__global__ void __launch_bounds__(256) k_inter(
        const float4* __restrict__ ent, const float* __restrict__ vals,
        const int* __restrict__ rows, const int* __restrict__ cols,
        float* __restrict__ out_user) {
    int tid = blockIdx.x * blockDim.x + threadIdx.x;
    int i   = tid >> 4;
    int sub = tid & 15;

    float v = vals[i];
    int rr = rows[i];
    int cc = cols[i];
    float4 ev = ent[cc * 16 + sub];
    int base = rr * CHANNEL + sub * 4;
    atomicAdd(&out_user[base + 0], v * ev.x);
    atomicAdd(&out_user[base + 1], v * ev.y);
    atomicAdd(&out_user[base + 2], v * ev.z);
    atomicAdd(&out_user[base + 3], v * ev.w);
}

// ---------------------------------------------------------------------------
// entity_agg = seg_sum / max(cnt, 1)   (in place on d_out entity region)
// ---------------------------------------------------------------------------
__global__ void k_div(float* __restrict__ out_ent, const float* __restrict__ cnt) {
    int idx = blockIdx.x * blockDim.x + threadIdx.x;   // N_ENTITIES*CHANNEL threads
    out_ent[idx] = out_ent[idx] / fmaxf(cnt[idx >> 6], 1.f);
}

// ---------------------------------------------------------------------------
// User gating with f32 WMMA: one wave (32 threads) per 16-user tile.
//   logits[16,16] = user_tile[16,64] @ latentT[64,16]   (latent padded 4->16)
//   via 16 chained V_WMMA_F32_16X16X4_F32 (exact f32 math, K=4 each).
// Then 4-wide softmax across lanes (xor 1,2 butterflies inside groups of 4),
// score broadcast via __shfl, gate = score @ disen in VALU (K=4 only),
// and in-place out_user *= (1 + gate).
//
// Layouts per ISA 7.12.2 (32-bit):
//   A 16x4 : lane l<16 -> M=l,    {K,K+1} in (a.x,a.y); l>=16 -> {K+2,K+3}
//   B 4x16 : lane l<16 -> N=l,    rows {K,K+1};          l>=16 -> rows {K+2,K+3}
//   D 16x16: VGPR r, lane l<16 -> D[r][l]; l>=16 -> D[r+8][l-16]
// ---------------------------------------------------------------------------
__global__ void __launch_bounds__(32) k_gate(
        const float* __restrict__ uemb, const float* __restrict__ lat,
        const float* __restrict__ disen, float* __restrict__ out_user) {
    const int lane = threadIdx.x;
    const int u0   = blockIdx.x * 16;          // 50000/16 = 3125 blocks exactly
    const int m    = lane & 15;
    const int koff = (lane < 16) ? 0 : 2;

    const float* arow = uemb + (u0 + m) * CHANNEL + koff;
    const float  bmask = (m < N_FACTORS) ? 1.f : 0.f;     // pad B columns 4..15 with 0
    const float* brow = lat + (m & 3) * CHANNEL + koff;   // B[k][n] = latent_emb[n][k]

    v8f acc = {0.f, 0.f, 0.f, 0.f, 0.f, 0.f, 0.f, 0.f};
#pragma unroll
    for (int k = 0; k < CHANNEL; k += 4) {
        v2f a, b;
        a.x = arow[k];
        a.y = arow[k + 1];
        b.x = brow[k]     * bmask;
        b.y = brow[k + 1] * bmask;
        acc = __builtin_amdgcn_wmma_f32_16x16x4_f32(
            /*neg_a=*/false, a, /*neg_b=*/false, b,
            /*c_mod=*/(short)0, acc, /*reuse_a=*/false, /*reuse_b=*/false);
    }

    // softmax over factor dim (n = 0..3). Row D[r][*] lives across lanes:
    // relevant groups {0..3} and {16..19} are xor-1/2 aligned.
    float sc[8];
#pragma unroll
    for (int r = 0; r < 8; ++r) {
        float x  = acc[r];
        float mx = x;
        mx = fmaxf(mx, __shfl_xor(mx, 1));
        mx = fmaxf(mx, __shfl_xor(mx, 2));
        float ex = __expf(x - mx);
        float s  = ex;
        s += __shfl_xor(s, 1);
        s += __shfl_xor(s, 2);
        sc[r] = ex / s;
    }

    // each lane owns 2 channels of the 64
    const int c = lane * 2;
    float d00 = disen[0 * CHANNEL + c], d01 = disen[0 * CHANNEL + c + 1];
    float d10 = disen[1 * CHANNEL + c], d11 = disen[1 * CHANNEL + c + 1];
    float d20 = disen[2 * CHANNEL + c], d21 = disen[2 * CHANNEL + c + 1];
    float d30 = disen[3 * CHANNEL + c], d31 = disen[3 * CHANNEL + c + 1];

#pragma unroll
    for (int r = 0; r < 8; ++r) {
        // user u0 + r : scores at lanes 0..3 of sc[r]
        float s0 = __shfl(sc[r], 0), s1 = __shfl(sc[r], 1);
        float s2 = __shfl(sc[r], 2), s3 = __shfl(sc[r], 3);
        float g0 = s0 * d00 + s1 * d10 + s2 * d20 + s3 * d30;
        float g1 = s0 * d01 + s1 * d11 + s2 * d21 + s3 * d31;
        int u = u0 + r;
        out_user[u * CHANNEL + c]     *= (1.f + g0);
        out_user[u * CHANNEL + c + 1] *= (1.f + g1);

        // user u0 + r + 8 : scores at lanes 16..19 of sc[r]
        float t0 = __shfl(sc[r], 16), t1 = __shfl(sc[r], 17);
        float t2 = __shfl(sc[r], 18), t3 = __shfl(sc[r], 19);
        g0 = t0 * d00 + t1 * d10 + t2 * d20 + t3 * d30;
        g1 = t0 * d01 + t1 * d11 + t2 * d21 + t3 * d31;
        u = u0 + r + 8;
        out_user[u * CHANNEL + c]     *= (1.f + g0);
        out_user[u * CHANNEL + c + 1] *= (1.f + g1);
    }
}

// ---------------------------------------------------------------------------
extern "C" void kernel_launch(void* const* d_in, const int* in_sizes, int n_in,
                              void* d_out, int out_size, void* d_ws, size_t ws_size,
                              hipStream_t stream) {
    const float* ent  = (const float*)d_in[0];   // entity_emb      [100000,64]
    const float* uemb = (const float*)d_in[1];   // user_emb        [50000,64]
    const float* lat  = (const float*)d_in[2];   // latent_emb      [4,64]
    const float* wgt  = (const float*)d_in[3];   // weight          [16,64]
    const float* datt = (const float*)d_in[4];   // disen_weight_att[4,16]
    const float* vals = (const float*)d_in[5];   // interact_vals   [1M]
    const int*   head = (const int*)d_in[6];
    const int*   tail = (const int*)d_in[7];
    const int*   etyp = (const int*)d_in[8];
    const int*   irow = (const int*)d_in[9];
    const int*   icol = (const int*)d_in[10];

    float* out      = (float*)d_out;
    float* out_ent  = out;                              // [100000*64]
    float* out_user = out + N_ENTITIES * CHANNEL;       // [ 50000*64]
    float* cnt      = (float*)d_ws;                     // [100000]
    float* disen    = cnt + N_ENTITIES;                 // [4*64]

    // 1) zero accumulators (fresh every call -> deterministic across replays)
    {
        int total = (N_ENTITIES + N_USERS) * CHANNEL + N_ENTITIES;
        k_zero<<<(total + 255) / 256, 256, 0, stream>>>(out, cnt);
    }
    // 2) tiny dense: disen = softmax(disen_weight_att) @ weight
    k_disen<<<1, 256, 0, stream>>>(datt, wgt, disen);
    // 3) edge scatter (1.6M * 16 lanes / 256 = 100000 blocks, exact)
    k_edges<<<N_EDGES * 16 / 256, 256, 0, stream>>>(
        (const float4*)ent, (const float4*)wgt, head, tail, etyp, out_ent, cnt);
    // 4) interaction scatter (1M * 16 / 256 = 62500 blocks, exact)
    k_inter<<<N_INTER * 16 / 256, 256, 0, stream>>>(
        (const float4*)ent, vals, irow, icol, out_user);
    // 5) scatter-mean divide
    k_div<<<N_ENTITIES * CHANNEL / 256, 256, 0, stream>>>(out_ent, cnt);
    // 6) WMMA gating path, one wave per 16 users (3125 waves, exact)
    k_gate<<<N_USERS / 16, 32, 0, stream>>>(uemb, lat, disen, out_user);
}